// MultiModelLinear_44152263803447
// MI455X (gfx1250) — compile-verified
//
#include <hip/hip_runtime.h>

typedef float v2f __attribute__((ext_vector_type(2)));
typedef float v8f __attribute__((ext_vector_type(8)));

#define NUM_MODELS 64
#define IN_F 256
#define OUT_F 256
#define BATCH 8192
#define MT 2                                   // M sub-tiles per wave (rows = 16*MT = 32)
#define NT 4                                   // N sub-tiles per wave (cols = 16*NT = 64)
#define MAX_TILES (BATCH / 32 + NUM_MODELS)    // 320: upper bound on sum(ceil(count_e/32))

// ---------------- grouping kernels ----------------

__global__ void mm_init(int* __restrict__ counts, int* __restrict__ cursor) {
    int t = threadIdx.x;
    if (t < NUM_MODELS) { counts[t] = 0; cursor[t] = 0; }
}

__global__ void mm_count(const long long* __restrict__ idx, int* __restrict__ counts) {
    int i = blockIdx.x * blockDim.x + threadIdx.x;
    if (i < BATCH) atomicAdd(&counts[(int)idx[i]], 1);
}

__global__ void mm_scan(const int* __restrict__ counts,
                        int* __restrict__ offsets, int* __restrict__ tileOff) {
    // single-thread exclusive scans (64 elements, trivial)
    int ro = 0, to = 0;
    offsets[0] = 0; tileOff[0] = 0;
    for (int e = 0; e < NUM_MODELS; ++e) {
        ro += counts[e];
        to += (counts[e] + 31) >> 5;           // 32-row tiles
        offsets[e + 1] = ro;
        tileOff[e + 1] = to;
    }
}

__global__ void mm_scatter(const long long* __restrict__ idx, int* __restrict__ cursor,
                           const int* __restrict__ offsets, int* __restrict__ perm) {
    int i = blockIdx.x * blockDim.x + threadIdx.x;
    if (i < BATCH) {
        int e = (int)idx[i];
        int pos = atomicAdd(&cursor[e], 1);
        perm[offsets[e] + pos] = i;
    }
}

// ---------------- grouped GEMM: one wave32 per 32x64 output tile ----------------
// D = A(32xK) * B(Kx64) + bias, K=256, via 512x V_WMMA_F32_16X16X4_F32 (exact f32).
// Each B operand is reused across MT=2 row tiles; each A operand across NT=4 col tiles.

__global__ __launch_bounds__(32) void mm_gemm(
    const float* __restrict__ x,       // [BATCH, IN_F]
    const float* __restrict__ w,       // [NUM_MODELS, OUT_F, IN_F] row-major
    const float* __restrict__ bias,    // [NUM_MODELS, OUT_F]
    const int* __restrict__ perm,      // grouped sample ids
    const int* __restrict__ offsets,   // [NUM_MODELS+1] row offsets
    const int* __restrict__ tileOff,   // [NUM_MODELS+1] 32-row-tile offsets
    float* __restrict__ out)           // [BATCH, OUT_F]
{
    const int tile = blockIdx.y;
    if (tile >= tileOff[NUM_MODELS]) return;

    // find owning expert (uniform scalar search over 64 entries)
    int e = 0;
    while (tileOff[e + 1] <= tile) ++e;

    const int rowBase = offsets[e] + (tile - tileOff[e]) * (16 * MT);
    const int rowLim  = offsets[e + 1];

    const int lane = threadIdx.x;        // 0..31
    const int half = lane >> 4;          // 0: K={0,1}, 1: K={2,3} within each 4-K step
    const int l15  = lane & 15;          // M for A, N for B

    // A row gathers through permutation (clamp out-of-range M to a valid row; not stored)
    const float* xrow[MT];
    #pragma unroll
    for (int mt = 0; mt < MT; ++mt) {
        const int myRow  = rowBase + mt * 16 + l15;
        const int srcRow = perm[(myRow < rowLim) ? myRow : rowBase];
        xrow[mt] = x + (size_t)srcRow * IN_F + half * 2;
    }

    const int nBase = blockIdx.x * (16 * NT);
    const float* wrow[NT];
    #pragma unroll
    for (int t = 0; t < NT; ++t)
        wrow[t] = w + ((size_t)e * OUT_F + nBase + t * 16 + l15) * IN_F + half * 2;

    v8f acc[MT][NT] = {};
    #pragma unroll 2
    for (int kk = 0; kk < IN_F / 4; ++kk) {
        // A 16x4 f32 per row tile: one float2 load, reused across NT col tiles
        v2f a[MT];
        #pragma unroll
        for (int mt = 0; mt < MT; ++mt)
            a[mt] = *(const v2f*)(xrow[mt] + kk * 4);
        #pragma unroll
        for (int t = 0; t < NT; ++t) {
            // B 4x16 f32 = W^T tile: one float2 load, reused across MT row tiles
            v2f bb = *(const v2f*)(wrow[t] + kk * 4);
            #pragma unroll
            for (int mt = 0; mt < MT; ++mt)
                acc[mt][t] = __builtin_amdgcn_wmma_f32_16x16x4_f32(
                    /*neg_a=*/false, a[mt], /*neg_b=*/false, bb,
                    /*c_mod=*/(short)0, acc[mt][t],
                    /*reuse_a=*/false, /*reuse_b=*/false);
        }
    }

    float bv[NT];
    #pragma unroll
    for (int t = 0; t < NT; ++t)
        bv[t] = bias[e * OUT_F + nBase + t * 16 + l15];

    // D layout: VGPR r holds (M = r + 8*half, N = l15); scatter rows back via perm
    #pragma unroll
    for (int mt = 0; mt < MT; ++mt) {
        #pragma unroll
        for (int r = 0; r < 8; ++r) {
            const int grow = rowBase + mt * 16 + r + 8 * half;
            if (grow < rowLim) {
                const int dstRow = perm[grow];   // broadcast load, uniform per 16-lane half
                float* orow = out + (size_t)dstRow * OUT_F + nBase + l15;
                #pragma unroll
                for (int t = 0; t < NT; ++t)
                    orow[t * 16] = acc[mt][t][r] + bv[t];
            }
        }
    }
}

// ---------------- launcher ----------------

extern "C" void kernel_launch(void* const* d_in, const int* in_sizes, int n_in,
                              void* d_out, int out_size, void* d_ws, size_t ws_size,
                              hipStream_t stream) {
    const float*     x    = (const float*)d_in[0];        // [8192, 256] f32
    const long long* idx  = (const long long*)d_in[1];    // [8192] int64
    const float*     w    = (const float*)d_in[2];        // [64, 256, 256] f32
    const float*     bias = (const float*)d_in[3];        // [64, 256] f32
    float*           out  = (float*)d_out;                // [8192, 256] f32

    int* ws      = (int*)d_ws;
    int* counts  = ws;            // 64
    int* cursor  = ws + 64;       // 64
    int* offsets = ws + 128;      // 65
    int* tileOff = ws + 256;      // 65
    int* perm    = ws + 384;      // 8192

    mm_init<<<1, 128, 0, stream>>>(counts, cursor);
    mm_count<<<BATCH / 256, 256, 0, stream>>>(idx, counts);
    mm_scan<<<1, 1, 0, stream>>>(counts, offsets, tileOff);
    mm_scatter<<<BATCH / 256, 256, 0, stream>>>(idx, cursor, offsets, perm);

    dim3 grid(OUT_F / (16 * NT), MAX_TILES);  // (4 col super-tiles, 320 tile slots)
    mm_gemm<<<grid, 32, 0, stream>>>(x, w, bias, perm, offsets, tileOff, out);
}